// CosFaceLoss_68247030333470
// MI455X (gfx1250) — compile-verified
//
#include <hip/hip_runtime.h>
#include <math.h>

// CosFace loss, fp32 end-to-end, fused one-pass softmax (fixed shift = 64
// since logits = 64*clip(cos,-1+eps,1-eps) < 64).
//
// W (100000 x 128, 51.2 MB) fits in MI455X's 192 MB L2; HBM sees W once.
// Compute uses V_WMMA_F32_16X16X4_F32 (full fp32 matrix path), two
// independent accumulator chains per wave, schedule pinned with
// sched_group_barrier to (2x ds_read -> 4x wmma) so both chains advance
// together with minimal B-fragment live range.
//
// SCALE=64 is folded into the LDS-staged normalized W (exact power-of-two
// scaling), so the epilogue clips in the scaled domain at +/-64*(1-1e-7),
// bit-equivalent to the reference's 64*clip(cos, -1+1e-7, 1-1e-7).

typedef __attribute__((ext_vector_type(2))) float v2f;
typedef __attribute__((ext_vector_type(8))) float v8f;

#define Bn 1024
#define Dn 128
#define Cn 100000
#define NC 128                       // classes per block tile
#define NCB ((Cn + NC - 1) / NC)     // 782 class blocks
#define SCALE_F 64.0f
#define MARGIN_S 22.4f               // 64 * 0.35
#define COS_HI ((float)(1.0 - 1e-7))
#define HI_S (64.0f * COS_HI)        // scaled-domain clip bounds (exact)
#define LO_S (-HI_S)

// workspace layout (float offsets)
#define WS_RNORM 0
#define WS_T     (WS_RNORM + Cn)     // target logits per row
#define WS_L     (WS_T + Bn)         // per-row losses
#define WS_SPART (WS_L + Bn)         // [NCB][Bn] partial exp-sums  (~3.2 MB)

// ---------------------------------------------------------------------------
// 1) rnorm[c] = 1 / max(||W[c,:]||, 1e-12); one wave32 per class row
// ---------------------------------------------------------------------------
__global__ __launch_bounds__(256) void rnorm_kernel(const float* __restrict__ W,
                                                    float* __restrict__ rnorm) {
  int c = (blockIdx.x * 256 + threadIdx.x) >> 5;   // wave index == class
  int lane = threadIdx.x & 31;
  if (c >= Cn) return;
  const float4* w4 = (const float4*)(W + (size_t)c * Dn);
  float4 v = w4[lane];                             // 32 lanes * 4 = 128 floats
  float s = v.x * v.x + v.y * v.y + v.z * v.z + v.w * v.w;
  #pragma unroll
  for (int m = 16; m >= 1; m >>= 1) s += __shfl_xor(s, m, 32);
  if (lane == 0) rnorm[c] = 1.0f / fmaxf(sqrtf(s), 1e-12f);
}

// ---------------------------------------------------------------------------
// 1b) target logit per row: T[b] = 64*clip(e_b . w_norm[label_b]) - 22.4
//     one wave32 per row; keeps conditional stores out of the hot GEMM loop
// ---------------------------------------------------------------------------
__global__ __launch_bounds__(256) void target_kernel(
    const float* __restrict__ E, const float* __restrict__ W,
    const int* __restrict__ labels, const float* __restrict__ rnorm,
    float* __restrict__ T) {
  int b = (blockIdx.x * 256 + threadIdx.x) >> 5;
  int lane = threadIdx.x & 31;
  if (b >= Bn) return;
  int lab = labels[b];
  const float4* e4 = (const float4*)(E + (size_t)b * Dn);
  const float4* w4 = (const float4*)(W + (size_t)lab * Dn);
  float4 e = e4[lane];
  float4 w = w4[lane];
  float s = e.x * w.x + e.y * w.y + e.z * w.z + e.w * w.w;
  #pragma unroll
  for (int m = 16; m >= 1; m >>= 1) s += __shfl_xor(s, m, 32);
  if (lane == 0) {
    float cosv = fminf(fmaxf(s * rnorm[lab], -COS_HI), COS_HI);
    T[b] = SCALE_F * cosv - MARGIN_S;
  }
}

// ---------------------------------------------------------------------------
// 2) main fused kernel: one block per 128-class tile; 8 waves x 8 b-tiles.
//    LDS holds 64*w_norm transposed as float2 (K-pair, class), XOR-swizzled
//    on pair parity so B-fragment ds_load_b64 is bank-conflict-free.
// ---------------------------------------------------------------------------
__global__ __launch_bounds__(256) void cosface_main(
    const float* __restrict__ E,
    const int* __restrict__ labels,
    const float* __restrict__ W,
    const float* __restrict__ rnorm,
    float* __restrict__ S_part) {
  __shared__ v2f sW[64 * NC];  // 64 KB: q = d/2 in 0..63, class 0..127 (swizzled)

  const int tid = threadIdx.x;
  const int c0 = blockIdx.x * NC;

  // ---- stage 64 * normalized W tile into LDS (transposed + swizzled) ----
  #pragma unroll 4
  for (int i = 0; i < (64 * NC) / 256; ++i) {      // 32 iterations
    int linear = tid + i * 256;
    int q = linear & 63;                           // K-pair index (d = 2q, 2q+1)
    int ci = linear >> 6;                          // class within tile
    int c = c0 + ci;
    v2f val = {0.0f, 0.0f};
    if (c < Cn) {
      const float* wr = W + (size_t)c * Dn + 2 * q;
      float rn = rnorm[c] * SCALE_F;               // fold logit scale (exact)
      val.x = wr[0] * rn;
      val.y = wr[1] * rn;
    }
    sW[q * NC + (ci ^ ((q & 1) << 4))] = val;
  }
  __syncthreads();

  const int wv = tid >> 5;       // wave 0..7
  const int lane = tid & 31;
  const int n = lane & 15;       // N / M position within fragment
  const int hi = lane >> 4;      // lane half

  #pragma unroll 1
  for (int bt = wv; bt < Bn / 16; bt += 8) {       // 8 b-tiles per wave
    const int b0 = bt * 16;

    // preload all 32 A-fragments for this b-tile (E rows b0..b0+15, fp32)
    // A 16x4 layout: lane m = lane&15; VGPR0/1 = K = 4*ks + 2*hi + {0,1}
    v2f a[32];
    const float* erow = E + (size_t)(b0 + n) * Dn;
    #pragma unroll
    for (int ks = 0; ks < 32; ++ks)
      a[ks] = *(const v2f*)(erow + ks * 4 + 2 * hi);

    // labels for the rows this lane's C-fragment touches: row = b0 + j + 8*hi
    int lab[8];
    #pragma unroll
    for (int j = 0; j < 8; ++j) lab[j] = labels[b0 + j + 8 * hi];

    float psum[8];
    #pragma unroll
    for (int j = 0; j < 8; ++j) psum[j] = 0.0f;

    #pragma unroll 1
    for (int ct = 0; ct < NC / 16; ct += 2) {      // 2 column sub-tiles / iter
      v8f acc0 = {};
      v8f acc1 = {};
      #pragma unroll
      for (int ks = 0; ks < 32; ++ks) {            // K = 128 in steps of 4
        int q = ks * 2 + hi;                       // B rows (2q, 2q+1)
        int sw = (q & 1) << 4;
        v2f bf0 = sW[q * NC + (((ct + 0) * 16 + n) ^ sw)];
        v2f bf1 = sW[q * NC + (((ct + 1) * 16 + n) ^ sw)];
        acc0 = __builtin_amdgcn_wmma_f32_16x16x4_f32(
            false, a[ks], false, bf0, (short)0, acc0, false, false);
        acc1 = __builtin_amdgcn_wmma_f32_16x16x4_f32(
            false, a[ks], false, bf1, (short)0, acc1, false, false);
      }
#if __has_builtin(__builtin_amdgcn_sched_group_barrier)
      // pin schedule: 2 merged 2addr DS reads feed 4 WMMAs (2 per chain);
      // both chains advance together, B-fragment live range stays tiny.
      #pragma unroll
      for (int i = 0; i < 16; ++i) {
        __builtin_amdgcn_sched_group_barrier(0x100, 2, 0);  // 2 DS reads
        __builtin_amdgcn_sched_group_barrier(0x008, 4, 0);  // 4 WMMA
      }
#endif
      // post-process both sub-tiles: D[m][n], m = j + 8*hi, n = lane&15
      if (c0 + NC <= Cn) {                         // uniform: full tile (781/782)
        #pragma unroll
        for (int half = 0; half < 2; ++half) {
          int cls = c0 + (ct + half) * 16 + n;
          #pragma unroll
          for (int j = 0; j < 8; ++j) {
            float v = half ? acc1[j] : acc0[j];
            float l = fminf(fmaxf(v, LO_S), HI_S); // scaled-domain clip
            float mar = (cls == lab[j]) ? MARGIN_S : 0.0f;
            psum[j] += __expf(l - mar - 64.0f);
          }
        }
      } else {                                     // tail tile: mask padding
        #pragma unroll
        for (int half = 0; half < 2; ++half) {
          int cls = c0 + (ct + half) * 16 + n;
          bool valid = cls < Cn;
          #pragma unroll
          for (int j = 0; j < 8; ++j) {
            float v = half ? acc1[j] : acc0[j];
            float l = fminf(fmaxf(v, LO_S), HI_S);
            float mar = (cls == lab[j]) ? MARGIN_S : 0.0f;
            float e = __expf(l - mar - 64.0f);
            psum[j] += valid ? e : 0.0f;
          }
        }
      }
    }

    // reduce per-row sums across the 16-lane N groups, store partials
    #pragma unroll
    for (int j = 0; j < 8; ++j) {
      float s = psum[j];
      s += __shfl_xor(s, 1, 32);
      s += __shfl_xor(s, 2, 32);
      s += __shfl_xor(s, 4, 32);
      s += __shfl_xor(s, 8, 32);
      if (n == j)
        S_part[(size_t)blockIdx.x * Bn + b0 + j + 8 * hi] = s;
    }
  }
}

// ---------------------------------------------------------------------------
// 3) per-row fixed-order reduction over class blocks -> per-row loss
// ---------------------------------------------------------------------------
__global__ __launch_bounds__(128) void reduce_rows(const float* __restrict__ S_part,
                                                   const float* __restrict__ T,
                                                   float* __restrict__ L) {
  int b = blockIdx.x * 128 + threadIdx.x;
  if (b >= Bn) return;
  float s = 0.0f;
  for (int j = 0; j < NCB; ++j) s += S_part[(size_t)j * Bn + b];  // coalesced
  L[b] = 64.0f + logf(s) - T[b];
}

// ---------------------------------------------------------------------------
// 4) mean over rows -> scalar
// ---------------------------------------------------------------------------
__global__ __launch_bounds__(256) void final_mean(const float* __restrict__ L,
                                                  float* __restrict__ out) {
  __shared__ float red[256];
  int t = threadIdx.x;
  red[t] = L[t] + L[t + 256] + L[t + 512] + L[t + 768];
  __syncthreads();
  for (int off = 128; off >= 1; off >>= 1) {
    if (t < off) red[t] += red[t + off];
    __syncthreads();
  }
  if (t == 0) out[0] = red[0] * (1.0f / 1024.0f);
}

extern "C" void kernel_launch(void* const* d_in, const int* in_sizes, int n_in,
                              void* d_out, int out_size, void* d_ws, size_t ws_size,
                              hipStream_t stream) {
  const float* E = (const float*)d_in[0];        // embeddings [1024,128] f32
  const int* labels = (const int*)d_in[1];       // labels [1024] int
  const float* W = (const float*)d_in[2];        // W [100000,128] f32

  float* ws = (float*)d_ws;                      // ~3.5 MB used
  float* rnorm = ws + WS_RNORM;
  float* T = ws + WS_T;
  float* L = ws + WS_L;
  float* S_part = ws + WS_SPART;

  rnorm_kernel<<<(Cn * 32 + 255) / 256, 256, 0, stream>>>(W, rnorm);
  target_kernel<<<(Bn * 32 + 255) / 256, 256, 0, stream>>>(E, W, labels, rnorm, T);
  cosface_main<<<NCB, 256, 0, stream>>>(E, labels, W, rnorm, S_part);
  reduce_rows<<<(Bn + 127) / 128, 128, 0, stream>>>(S_part, T, L);
  final_mean<<<1, 256, 0, stream>>>(L, (float*)d_out);
}